// DeepGNN_17686675325464
// MI455X (gfx1250) — compile-verified
//
#include <hip/hip_runtime.h>
#include <hip/hip_bf16.h>
#include <stdint.h>

typedef float v2f __attribute__((ext_vector_type(2)));
typedef float v8f __attribute__((ext_vector_type(8)));
typedef unsigned int u32x4 __attribute__((ext_vector_type(4)));
typedef int i32x4 __attribute__((ext_vector_type(4)));
typedef int i32x8 __attribute__((ext_vector_type(8)));

#define HID 256

// ---------------- degree / normalization ----------------

__global__ void init_deg_kernel(float* deg, int n) {
    int i = blockIdx.x * blockDim.x + threadIdx.x;
    if (i < n) deg[i] = 1.0f;  // self-loop contributes 1 to every node's degree
}

__global__ void deg_edges_kernel(const long long* __restrict__ dst, float* deg, long long E) {
    long long e = (long long)blockIdx.x * blockDim.x + threadIdx.x;
    if (e < E) atomicAdd(&deg[(int)dst[e]], 1.0f);
}

__global__ void finalize_dinv_kernel(float* deg, int n) {
    int i = blockIdx.x * blockDim.x + threadIdx.x;
    if (i < n) {
        float d = deg[i];
        deg[i] = (d > 0.0f) ? rsqrtf(d) : 0.0f;
    }
}

// ---------------- dense GEMM via f32 WMMA + TDM-staged A tile ----------------
// C[M,N] = A[M,K] @ W[K,N], row-major f32. One 512-thread block (16 waves) owns a
// 16-row M strip; wave w computes N-tile w. The 16xK A strip is staged to LDS once
// (TDM with row padding of 4 dwords -> LDS stride K+4, bank-conflict-free ds reads).
// Requires M%16==0 (50000=3125*16), N==256, K in {128,256}.
__global__ __launch_bounds__(512) void gemm_wmma_f32(const float* __restrict__ A,
                                                     const float* __restrict__ W,
                                                     float* __restrict__ out,
                                                     int M, int K, int N) {
    __shared__ float As[16 * (256 + 4)];   // max K=256, padded stride K+4

    const int row0 = blockIdx.x << 4;
    if (row0 >= M) return;
    const int strideA = K + 4;

#if defined(__has_builtin) && __has_builtin(__builtin_amdgcn_tensor_load_to_lds)
    if (threadIdx.x < 32) {
        // Build Tensor DMA Descriptor (D#) per CDNA5 ISA.
        const unsigned lds_off = (unsigned)(uintptr_t)(void*)&As[0];
        const unsigned long long ga = (unsigned long long)(uintptr_t)(A + (size_t)row0 * K);
        // Group 0: count=1 | lds_addr | global_addr(57b) | type=2
        u32x4 g0;
        g0.x = 1u;                                   // count=1, is_restore=0, gather off
        g0.y = lds_off;                              // LDS byte address
        g0.z = (unsigned)(ga & 0xFFFFFFFFull);       // global_addr[31:0]
        g0.w = (unsigned)((ga >> 32) & 0x01FFFFFFull) | (2u << 30);  // [56:32] | type=2
        // Group 1: data_size=4B, pad_enable, pad_interval=K dwords, pad_amount=4 dwords,
        // tensor = [16 rows x K], tile = [K x 16 rows], row stride = K.
        const int padIv = (K == 128) ? 6 : 7;        // 2<<v dwords: 6->128, 7->256
        i32x8 g1;
        g1[0] = (2 << 16) | (1 << 20) | (padIv << 22) | (3 << 25);  // pad_amount code 3 -> 4 dwords
        g1[1] = (K & 0xFFFF) << 16;                  // tensor_dim0[15:0]
        g1[2] = ((K >> 16) & 0xFFFF) | (16 << 16);   // tensor_dim0[31:16] | tensor_dim1=16
        g1[3] = (K & 0xFFFF) << 16;                  // tensor_dim1 hi=0 | tile_dim0=K
        g1[4] = 16;                                  // tile_dim1=16, tile_dim2=0
        g1[5] = K;                                   // tensor_dim0_stride[31:0]
        g1[6] = 0;
        g1[7] = 0;
        i32x4 z4 = {0, 0, 0, 0};                     // groups 2/3 unused (2D tensor)
        i32x8 z8 = {0, 0, 0, 0, 0, 0, 0, 0};         // 6-arg toolchain form: extra group
        __builtin_amdgcn_tensor_load_to_lds(g0, g1, z4, z4, z8, 0);
        __builtin_amdgcn_s_wait_tensorcnt(0);
    }
#else
    for (int idx = threadIdx.x; idx < 16 * K; idx += blockDim.x) {
        int r = idx / K, cc = idx - r * K;
        As[r * strideA + cc] = A[(size_t)(row0 + r) * K + cc];
    }
#endif
    __syncthreads();

    const int nt   = threadIdx.x >> 5;     // wave id = N tile, 0..15
    const int lane = threadIdx.x & 31;
    const int col0 = nt << 4;
    const int m    = lane & 15;            // M (for A) / N (for B) index within tile
    const int koff = (lane >> 4) << 1;     // lanes 16-31 carry K+2,K+3 per ISA layout

    const float* abase = &As[m * strideA + koff];             // LDS, 8B-aligned reads
    const float* bcol  = W + (size_t)koff * N + col0 + m;     // global, L2-resident

    v8f c = {};
    #pragma unroll 8
    for (int k = 0; k < K; k += 4) {
        v2f a = *(const v2f*)(abase + k);                     // ds_load_b64, conflict-free
        v2f b;
        b.x = bcol[(size_t)k * N];
        b.y = bcol[(size_t)(k + 1) * N];
        c = __builtin_amdgcn_wmma_f32_16x16x4_f32(
                false, a, false, b, (short)0, c, false, false);
    }

    // D layout: lanes 0-15 -> rows M=0..7 (vgpr v), lanes 16-31 -> rows M=8..15
    const int rbase = row0 + ((lane >> 4) << 3);
    const int cidx  = col0 + m;
    #pragma unroll
    for (int v = 0; v < 8; ++v)
        out[(size_t)(rbase + v) * N + cidx] = c[v];
}

// ---------------- GCN aggregation ----------------

__global__ void self_loop_init_kernel(const float* __restrict__ tmp,
                                      const float* __restrict__ dinv,
                                      float* __restrict__ hout) {
    const long long i = blockIdx.x;
    const int c = threadIdx.x;
    const float di = dinv[i];
    hout[i * HID + c] = tmp[i * HID + c] * di * di;
}

__global__ void edge_scatter_kernel(const long long* __restrict__ src,
                                    const long long* __restrict__ dst,
                                    const float* __restrict__ dinv,
                                    const float* __restrict__ tmp,
                                    float* __restrict__ hout) {
    const long long e = blockIdx.x;
    const int c = threadIdx.x;
    const int s = (int)src[e];
    const int d = (int)dst[e];
    const float w = dinv[s] * dinv[d];
    const float val = tmp[(long long)s * HID + c] * w;
    atomicAdd(&hout[(long long)d * HID + c], val);
}

__global__ void bias_relu_kernel(float* __restrict__ h, const float* __restrict__ b) {
    const long long i = blockIdx.x;
    const int c = threadIdx.x;
    float v = h[i * HID + c] + b[c];
    h[i * HID + c] = v > 0.0f ? v : 0.0f;
}

// ---------------- pooling + final linear ----------------

__global__ void pool_init_kernel(float* pooled, float* cnt, int nG) {
    int i = blockIdx.x * blockDim.x + threadIdx.x;
    if (i < nG * HID) pooled[i] = 0.0f;
    if (i < nG) cnt[i] = 0.0f;
}

__global__ void pool_accum_kernel(const float* __restrict__ h,
                                  const long long* __restrict__ batch,
                                  float* __restrict__ pooled,
                                  float* __restrict__ cnt) {
    const long long i = blockIdx.x;
    const int c = threadIdx.x;
    const int g = (int)batch[i];
    atomicAdd(&pooled[(long long)g * HID + c], h[i * HID + c]);
    if (c == 0) atomicAdd(&cnt[g], 1.0f);
}

__global__ void final_linear_kernel(const float* __restrict__ pooled,
                                    const float* __restrict__ cnt,
                                    const float* __restrict__ Wlin,
                                    const float* __restrict__ blin,
                                    float* __restrict__ out, int nG, int nOut) {
    int idx = blockIdx.x * blockDim.x + threadIdx.x;
    if (idx >= nG * nOut) return;
    const int g = idx / nOut;
    const int j = idx - g * nOut;
    const float inv = 1.0f / fmaxf(cnt[g], 1.0f);
    float acc = 0.0f;
    for (int c = 0; c < HID; ++c)
        acc += pooled[g * HID + c] * Wlin[c * nOut + j];
    out[idx] = acc * inv + blin[j];
}

// ---------------- launch ----------------

extern "C" void kernel_launch(void* const* d_in, const int* in_sizes, int n_in,
                              void* d_out, int out_size, void* d_ws, size_t ws_size,
                              hipStream_t stream) {
    const float*     x    = (const float*)d_in[0];
    const long long* ei   = (const long long*)d_in[1];
    const long long* batch= (const long long*)d_in[2];
    const float* W1 = (const float*)d_in[3];  const float* b1 = (const float*)d_in[4];
    const float* W2 = (const float*)d_in[5];  const float* b2 = (const float*)d_in[6];
    const float* W3 = (const float*)d_in[7];  const float* b3 = (const float*)d_in[8];
    const float* W4 = (const float*)d_in[9];  const float* b4 = (const float*)d_in[10];
    const float* Wl = (const float*)d_in[11]; const float* bl = (const float*)d_in[12];
    float* out = (float*)d_out;

    const int       N   = in_sizes[0] / 128;      // 50000 nodes
    const int       K0  = 128;
    const long long E   = in_sizes[1] / 2;        // 1.6M edges
    const int       nG  = 256;
    const int       nOut= 10;

    const long long* src = ei;
    const long long* dst = ei + E;

    // workspace layout (floats)
    float* bufA   = (float*)d_ws;                 // [N, HID]
    float* bufB   = bufA + (long long)N * HID;    // [N, HID]
    float* dinv   = bufB + (long long)N * HID;    // [N]
    float* pooled = dinv + N;                     // [nG, HID]
    float* cnt    = pooled + (long long)nG * HID; // [nG]

    // --- degree / dinv ---
    init_deg_kernel<<<(N + 255) / 256, 256, 0, stream>>>(dinv, N);
    deg_edges_kernel<<<(int)((E + 255) / 256), 256, 0, stream>>>(dst, dinv, E);
    finalize_dinv_kernel<<<(N + 255) / 256, 256, 0, stream>>>(dinv, N);

    const int mBlocks = N / 16;                   // 3125 blocks, 16 waves each

    // --- layer 1: x(K=128) -> bufA ---
    gemm_wmma_f32<<<mBlocks, 512, 0, stream>>>(x, W1, bufB, N, K0, HID);
    self_loop_init_kernel<<<N, HID, 0, stream>>>(bufB, dinv, bufA);
    edge_scatter_kernel<<<(int)E, HID, 0, stream>>>(src, dst, dinv, bufB, bufA);
    bias_relu_kernel<<<N, HID, 0, stream>>>(bufA, b1);

    // --- layers 2..4 ---
    const float* Ws[3] = {W2, W3, W4};
    const float* bs[3] = {b2, b3, b4};
    for (int l = 0; l < 3; ++l) {
        gemm_wmma_f32<<<mBlocks, 512, 0, stream>>>(bufA, Ws[l], bufB, N, HID, HID);
        self_loop_init_kernel<<<N, HID, 0, stream>>>(bufB, dinv, bufA);
        edge_scatter_kernel<<<(int)E, HID, 0, stream>>>(src, dst, dinv, bufB, bufA);
        bias_relu_kernel<<<N, HID, 0, stream>>>(bufA, bs[l]);
    }

    // --- mean pool + final linear ---
    pool_init_kernel<<<(nG * HID + 255) / 256, 256, 0, stream>>>(pooled, cnt, nG);
    pool_accum_kernel<<<N, HID, 0, stream>>>(bufA, batch, pooled, cnt);
    final_linear_kernel<<<(nG * nOut + 255) / 256, 256, 0, stream>>>(pooled, cnt, Wl, bl, out, nG, nOut);
}